// GraphSAGEEncoder_24687472017932
// MI455X (gfx1250) — compile-verified
//
#include <hip/hip_runtime.h>

typedef __attribute__((ext_vector_type(2))) float v2f;
typedef __attribute__((ext_vector_type(4))) float v4f;
typedef __attribute__((ext_vector_type(8))) float v8f;

#define NNODES 100000
#define DIM 128
#define NSAMP 10
#define LDSW_STRIDE 144   // 128 cols + 16 pad (float2) => halves of wave hit disjoint LDS banks

// ---------------------------------------------------------------------------
// Neighbor mean: one wave per node, one lane per 4 features (coalesced 512B rows).
// out[n, :] = (1/S) * sum_s h[idx[n,s], :]
// ---------------------------------------------------------------------------
__global__ __launch_bounds__(256) void sage_mean(const float* __restrict__ h,
                                                 const int* __restrict__ idx,
                                                 float* __restrict__ out) {
  int gid = blockIdx.x * blockDim.x + threadIdx.x;
  int n = gid >> 5;                 // wave-uniform node id
  if (n >= NNODES) return;
  int lane = gid & 31;
  const int* __restrict__ nb = idx + n * NSAMP;
  v4f acc = {0.f, 0.f, 0.f, 0.f};
#pragma unroll
  for (int s = 0; s < NSAMP; ++s) {
    int j = nb[s];
    v4f v = *(const v4f*)(h + (long)j * DIM + lane * 4);
    acc += v;
  }
  acc *= 0.1f;
  *(v4f*)(out + (long)n * DIM + lane * 4) = acc;
}

// ---------------------------------------------------------------------------
// Y = relu([h | mean] @ W^T + b) using V_WMMA_F32_16X16X4_F32 (exact fp32).
// W: [128, 256] row-major.  Block = 8 waves; wave w computes rows
// [tile*16, tile*16+16) x all 128 output cols (8 col-tiles, v8f each).
// LDS holds W as pairs: ldsW[p*STRIDE + o] = (W[o,2p], W[o,2p+1]).
// ---------------------------------------------------------------------------
__global__ __launch_bounds__(256) void sage_gemm(const float* __restrict__ hin,
                                                 const float* __restrict__ mean,
                                                 const float* __restrict__ W,
                                                 const float* __restrict__ bias,
                                                 float* __restrict__ out) {
  __shared__ v2f ldsW[128 * LDSW_STRIDE];   // 144 KB

  const int tid = threadIdx.x;
  // Cooperative W load: consecutive threads -> consecutive k-pairs (coalesced global).
  for (int e = tid; e < 128 * 128; e += 256) {
    int p = e & 127;       // k-pair index (k = 2p, 2p+1)
    int o = e >> 7;        // output column
    ldsW[p * LDSW_STRIDE + o] = *(const v2f*)(W + o * 256 + 2 * p);
  }
  __syncthreads();

  const int lane = tid & 31;
  const int wave = tid >> 5;
  const int tile = blockIdx.x * 8 + wave;   // 16-row tile (wave-uniform)
  const int ntiles = NNODES / 16;           // 6250, exact
  if (tile >= ntiles) return;               // whole wave exits -> EXEC stays all-1s

  const int m   = lane & 15;                // A row / B col within tile
  const int khi = lane >> 4;                // 0: K={0,1}, 1: K={2,3}
  const long rowA = (long)tile * 16 + m;
  const float* __restrict__ hrow = hin  + rowA * DIM;
  const float* __restrict__ mrow = mean + rowA * DIM;

  v8f acc[8];
#pragma unroll
  for (int c = 0; c < 8; ++c) acc[c] = (v8f){0.f,0.f,0.f,0.f,0.f,0.f,0.f,0.f};

#pragma unroll 2
  for (int kk = 0; kk < 256; kk += 4) {
    // A fragment (16x4 f32 layout): lane holds cat[rowA, kbase], cat[rowA, kbase+1]
    const float* src = (kk < 128) ? hrow : mrow;     // uniform select per kk
    int kloc = (kk & 127) + khi * 2;
    v2f a = *(const v2f*)(src + kloc);
    int prow = (kk >> 1) + khi;                      // pair row in ldsW
#pragma unroll
    for (int c = 0; c < 8; ++c) {
      // B fragment (4x16 f32): lane holds B[kbase, col], B[kbase+1, col] = W[col, kbase..+1]
      v2f b = ldsW[prow * LDSW_STRIDE + c * 16 + m];
      acc[c] = __builtin_amdgcn_wmma_f32_16x16x4_f32(
          /*neg_a=*/false, a, /*neg_b=*/false, b,
          /*c_mod=*/(short)0, acc[c], /*reuse_a=*/false, /*reuse_b=*/false);
    }
  }

  // Epilogue: bias + relu + store (C/D layout: VGPR j -> M = j + khi*8, N = m)
#pragma unroll
  for (int c = 0; c < 8; ++c) {
    int col = c * 16 + m;
    float bv = bias[col];
#pragma unroll
    for (int j = 0; j < 8; ++j) {
      long row = (long)tile * 16 + khi * 8 + j;
      float v = acc[c][j] + bv;
      out[row * DIM + col] = v > 0.f ? v : 0.f;
    }
  }
}

// ---------------------------------------------------------------------------
extern "C" void kernel_launch(void* const* d_in, const int* in_sizes, int n_in,
                              void* d_out, int out_size, void* d_ws, size_t ws_size,
                              hipStream_t stream) {
  const float* h   = (const float*)d_in[0];
  const int*  idx1 = (const int*)d_in[1];
  const int*  idx2 = (const int*)d_in[2];
  const float* W1  = (const float*)d_in[3];
  const float* b1  = (const float*)d_in[4];
  const float* W2  = (const float*)d_in[5];
  const float* b2  = (const float*)d_in[6];
  float* out = (float*)d_out;
  float* meanbuf = (float*)d_ws;            // N*D floats = 51.2 MB scratch

  dim3 meanGrid((NNODES * 32 + 255) / 256); // one wave per node
  dim3 gemmGrid((NNODES / 16 + 7) / 8);     // 8 row-tiles per block

  // Layer 1: hidden lives in d_out
  sage_mean<<<meanGrid, 256, 0, stream>>>(h, idx1, meanbuf);
  sage_gemm<<<gemmGrid, 256, 0, stream>>>(h, meanbuf, W1, b1, out);
  // Layer 2: mean precomputed, gemm reads/writes disjoint-per-wave rows of d_out
  sage_mean<<<meanGrid, 256, 0, stream>>>(out, idx2, meanbuf);
  sage_gemm<<<gemmGrid, 256, 0, stream>>>(out, meanbuf, W2, b2, out);
}